// BinnedColorLoss_55430847922669
// MI455X (gfx1250) — compile-verified
//
#include <hip/hip_runtime.h>
#include <stdint.h>

// Problem constants (B, C, H, W, K) = (16, 313, 128, 128, 5)
#define B_   16
#define C_   313
#define H_   128
#define W_   128
#define K_   5
#define HW_  (H_ * W_)            // 16384
#define NPIX_ (B_ * HW_)          // 262144
#define TPB  256                  // 8 wave32s
#define NBLK (NPIX_ / TPB)        // 1024 blocks
#define CCH  16                   // channels per TDM tile
#define NCH  ((C_ + CCH - 1) / CCH) // 20 chunks (last has 9 valid channels)

static_assert(NPIX_ % TPB == 0, "pixel count must tile");
static_assert(HW_ % TPB == 0, "block stays within one batch image");

typedef unsigned int v4u __attribute__((ext_vector_type(4)));
typedef int          v8i __attribute__((ext_vector_type(8)));
typedef int          v4i __attribute__((ext_vector_type(4)));

struct SMem {
  float buf[2][CCH * TPB];  // 2 x 16KB double-buffered TDM tiles
  float red[TPB];           // block reduction scratch
};

// Issue one TDM 2D tile load: tile_dim0 = 256 contiguous floats (pixels),
// tile_dim1 = ccur channels, channel stride = HW_ elements.
// Descriptor layout per CDNA5 ISA.
__device__ __forceinline__ void tdm_load_tile(uint32_t lds_off,
                                              const float* gptr, int ccur) {
  uint64_t ga = (uint64_t)(uintptr_t)gptr;
  v4u g0;
  g0[0] = 1u;                                    // count=1, user mode
  g0[1] = lds_off;                               // lds_addr (bytes)
  g0[2] = (uint32_t)(ga & 0xffffffffu);          // global_addr[31:0]
  g0[3] = (uint32_t)((ga >> 32) & 0x1ffffffu)    // global_addr[56:32]
        | (2u << 30);                            // type = 2 ("image")
  v8i g1;
  g1[0] = (int)(2u << 16);                       // wg_mask=0, data_size=2 (4B)
  g1[1] = (int)((uint32_t)TPB << 16);            // tensor_dim0[15:0] = 256
  g1[2] = (int)(((uint32_t)ccur & 0xffffu) << 16); // tensor_dim1[15:0] = ccur
  g1[3] = (int)((uint32_t)TPB << 16);            // tile_dim0 = 256
  g1[4] = (int)(((uint32_t)ccur & 0xffffu) | (1u << 16)); // tile_dim1=ccur, tile_dim2=1
  g1[5] = (int)(uint32_t)HW_;                    // tensor_dim0_stride = 16384
  g1[6] = 0;                                     // stride hi / dim1_stride lo
  g1[7] = 0;                                     // dim1_stride hi (z unused)
  v4i g2;                                        // tensor_dim2=1, tensor_dim3=1
  g2[0] = 1; g2[1] = 1; g2[2] = 0; g2[3] = 0;    // tile_dim3=0 (unused)
  v4i g3;                                        // tensor_dim4=1, tile_dim4=0
  g3[0] = 0; g3[1] = (int)(1u << 16); g3[2] = 0; g3[3] = 0;
  v8i g4 = {};                                   // extra descriptor words: zero
  __builtin_amdgcn_tensor_load_to_lds(g0, g1, g2, g3, g4, 0);
}

__global__ __launch_bounds__(TPB) void binned_loss_main(
    const float* __restrict__ pred,       // (B, C, H, W)
    const int*   __restrict__ binned,     // (B, 1, H, W)
    const int*   __restrict__ knn_idx,    // (C, K)
    const float* __restrict__ knn_wts,    // (C, K)
    const float* __restrict__ weights,    // (C,)
    float*       __restrict__ partial)    // (NBLK,)
{
  __shared__ SMem sm;
  const int tid = threadIdx.x;
  const int blk = blockIdx.x;
  const int p   = blk * TPB + tid;        // global pixel id
  const int b   = blk >> 6;               // 64 blocks per HW plane
  const int hw0 = (blk & 63) * TPB;       // pixel offset of lane 0 in plane

  // Per-pixel lookup tables (tiny, L2-resident)
  const int t = binned[p];
  int   kidx[K_];
  float kwt[K_];
  float wsum = 0.f;
#pragma unroll
  for (int k = 0; k < K_; ++k) {
    kidx[k] = knn_idx[t * K_ + k];
    kwt[k]  = knn_wts[t * K_ + k];
    wsum   += kwt[k];
  }
  const float wpix = weights[t];

  const float* gbase = pred + ((size_t)b * C_) * HW_ + hw0;
  const bool issuer = (tid < 32);  // wave 0 drives the TDM (EXEC ignored)

  // Flat shared pointer low 32 bits == workgroup-relative LDS byte offset
  const uint32_t lds0 = (uint32_t)(uintptr_t)&sm.buf[0][0];
  const uint32_t lds1 = (uint32_t)(uintptr_t)&sm.buf[1][0];

  if (issuer) {
    tdm_load_tile(lds0, gbase, CCH);                 // chunk 0 -> buf0
    __builtin_amdgcn_s_wait_tensorcnt(0);
  }
  __syncthreads();                                   // buf0 visible to all

  float m = -3.0e38f;   // running max (online logsumexp)
  float s = 0.0f;       // running scaled sum
  float dot = 0.0f;     // Σ_k wts_k * pred[idx_k]

  for (int ch = 0; ch < NCH; ++ch) {
    const int c0   = ch * CCH;
    const int ccur = (C_ - c0 < CCH) ? (C_ - c0) : CCH;
    const int cur  = ch & 1;

    // Prefetch next chunk into the other buffer (readers of it passed the
    // end-of-iteration barrier two barriers ago -> WAR safe).
    if (issuer && (ch + 1 < NCH)) {
      const int nc0  = c0 + CCH;
      const int ncur = (C_ - nc0 < CCH) ? (C_ - nc0) : CCH;
      tdm_load_tile(cur ? lds0 : lds1, gbase + (size_t)nc0 * HW_, ncur);
    }

    const float* bufc = &sm.buf[cur][0];

    // Chunk max (registers), then chunk Σexp, then online merge: ~1 exp/elem
    float rv[CCH];
    float cm = -3.0e38f;
#pragma unroll
    for (int cc = 0; cc < CCH; ++cc) {
      float v = (cc < ccur) ? bufc[cc * TPB + tid] : -3.0e38f;
      rv[cc] = v;
      cm = fmaxf(cm, v);
    }
    float cs = 0.0f;
#pragma unroll
    for (int cc = 0; cc < CCH; ++cc) cs += __expf(rv[cc] - cm);

    const float nm = fmaxf(m, cm);
    s = s * __expf(m - nm) + cs * __expf(cm - nm);
    m = nm;

    // Service the K-NN gather from the in-flight LDS tile: no extra HBM traffic
#pragma unroll
    for (int k = 0; k < K_; ++k) {
      const int rel = kidx[k] - c0;
      if ((unsigned)rel < (unsigned)ccur)
        dot += kwt[k] * bufc[rel * TPB + tid];
    }

    __syncthreads();                                 // all reads of cur done
    if (ch + 1 < NCH) {
      if (issuer) __builtin_amdgcn_s_wait_tensorcnt(0);
      __syncthreads();                               // next buffer visible
    }
  }

  const float lse = m + __logf(s);
  float contrib = (dot - wsum * lse) * wpix;         // Σ_k wts_k*logp[idx_k]*w

  // Block tree reduction
  sm.red[tid] = contrib;
  __syncthreads();
  for (int off = TPB / 2; off > 0; off >>= 1) {
    if (tid < off) sm.red[tid] += sm.red[tid + off];
    __syncthreads();
  }
  if (tid == 0) partial[blk] = sm.red[0];
}

__global__ __launch_bounds__(TPB) void binned_loss_final(
    const float* __restrict__ partial, float* __restrict__ out) {
  __shared__ float red[TPB];
  float v = 0.f;
  for (int i = threadIdx.x; i < NBLK; i += TPB) v += partial[i];
  red[threadIdx.x] = v;
  __syncthreads();
  for (int off = TPB / 2; off > 0; off >>= 1) {
    if (threadIdx.x < off) red[threadIdx.x] += red[threadIdx.x + off];
    __syncthreads();
  }
  if (threadIdx.x == 0) out[0] = -red[0] / (float)NPIX_;
}

extern "C" void kernel_launch(void* const* d_in, const int* in_sizes, int n_in,
                              void* d_out, int out_size, void* d_ws, size_t ws_size,
                              hipStream_t stream) {
  (void)in_sizes; (void)n_in; (void)out_size; (void)ws_size;
  const float* pred    = (const float*)d_in[0];
  // d_in[1] (_color) is unused by the reference loss
  const int*   binned  = (const int*)d_in[2];
  const int*   knn_idx = (const int*)d_in[3];
  const float* knn_wts = (const float*)d_in[4];
  const float* weights = (const float*)d_in[5];
  float* partial = (float*)d_ws;   // needs NBLK*4 = 4 KB of scratch
  float* out     = (float*)d_out;

  binned_loss_main<<<dim3(NBLK), dim3(TPB), 0, stream>>>(
      pred, binned, knn_idx, knn_wts, weights, partial);
  binned_loss_final<<<dim3(1), dim3(TPB), 0, stream>>>(partial, out);
}